// CombineGraph_26886495272978
// MI455X (gfx1250) — compile-verified
//
#include <hip/hip_runtime.h>
#include <hip/hip_bf16.h>
#include <math.h>

#define B_ 128
#define L_ 50
#define D_ 128
#define S_ 8
#define NN_ 100000
#define KP_ 132          // 129 padded to multiple of 4
#define LEAKY_ 0.2f
#define BETA_ 0.005f

typedef float v2f __attribute__((ext_vector_type(2)));
typedef float v8f __attribute__((ext_vector_type(8)));

#if defined(__has_builtin)
#if __has_builtin(__builtin_amdgcn_wmma_f32_16x16x4_f32)
#define HAVE_WMMA_F32 1
#endif
#endif

__device__ __forceinline__ v8f wmma4(v2f a, v2f b, v8f c) {
#ifdef HAVE_WMMA_F32
  return __builtin_amdgcn_wmma_f32_16x16x4_f32(false, a, false, b, (short)0, c,
                                               false, false);
#else
  float* cf = (float*)&c;
  #pragma unroll
  for (int i = 0; i < 8; ++i) cf[i] += a.x * b.x + a.y * b.y;  // placeholder
  return c;
#endif
}

__device__ __forceinline__ float leaky_(float x) { return x > 0.f ? x : LEAKY_ * x; }
__device__ __forceinline__ float sigmoid_(float x) { return 1.f / (1.f + __expf(-x)); }

// ---------------------------------------------------------------- prep -------
// gw1T[hop][n][k] = gw1[hop][k][n] (k<129) else 0   -> [2][128][KP_]
__global__ void gw1t_kernel(const float* __restrict__ gw1, float* __restrict__ gw1T) {
  int i = blockIdx.x * 256 + threadIdx.x;
  if (i >= 2 * D_ * KP_) return;
  int hop = i / (D_ * KP_);
  int rem = i % (D_ * KP_);
  int n = rem / KP_, k = rem % KP_;
  gw1T[i] = (k < 129) ? gw1[hop * 129 * D_ + k * D_ + n] : 0.f;
}

// w1botT[n][k] = w1[128+k][n]  -> [128][128]
__global__ void w1t_kernel(const float* __restrict__ w1, float* __restrict__ w1botT) {
  int i = blockIdx.x * 256 + threadIdx.x;
  if (i >= D_ * D_) return;
  int n = i >> 7, k = i & 127;
  w1botT[i] = w1[(D_ + k) * D_ + n];
}

// ---------------------------------------------------------------- gathers ----
__global__ void gather_h_kernel(const float* __restrict__ emb,
                                const int* __restrict__ inputs,
                                float* __restrict__ h, int total) {
  int i = blockIdx.x * 256 + threadIdx.x;
  if (i >= total) return;
  int row = i >> 7, d = i & 127;
  h[i] = emb[((long)inputs[row] << 7) + d];
}

__global__ void hop_kernel(const int* __restrict__ src,
                           const int* __restrict__ adj_all,
                           const float* __restrict__ num_w,
                           int* __restrict__ dst, float* __restrict__ dstw,
                           int total) {
  int i = blockIdx.x * 256 + threadIdx.x;
  if (i >= total) return;
  int node = src[i >> 3];
  int s = i & 7;
  dst[i]  = adj_all[node * S_ + s];
  dstw[i] = num_w[node * S_ + s];
}

__global__ void sess_kernel(const float* __restrict__ emb,
                            const int* __restrict__ item,
                            const int* __restrict__ mask,
                            float* __restrict__ sess) {
  int b = blockIdx.x, d = threadIdx.x;
  float acc = 0.f, ms = 0.f;
  for (int l = 0; l < L_; ++l) {
    float m = (float)mask[b * L_ + l];
    ms += m;
    acc += emb[((long)item[b * L_ + l] << 7) + d] * m;
  }
  sess[(b << 7) + d] = acc / ms;
}

// ------------------------------------------------------------- local agg ----
__global__ void local_agg_kernel(const float* __restrict__ h,
                                 const int* __restrict__ adj,
                                 const float* __restrict__ a_local,
                                 float* __restrict__ h_local) {
  int b = blockIdx.x;
  __shared__ float hv[L_][D_];
  __shared__ float att[L_][L_];
  __shared__ float rinv[L_];
  for (int i = threadIdx.x; i < L_ * D_; i += 256)
    hv[i / D_][i & 127] = h[(long)b * L_ * D_ + i];
  __syncthreads();
  for (int p = threadIdx.x; p < L_ * L_; p += 256) {
    int i = p / L_, j = p % L_;
    int k = adj[b * L_ * L_ + p];
    float v;
    if (k == 0) v = -9e15f;
    else {
      const float* a = a_local + (k - 1) * D_;
      float acc = 0.f;
      for (int d = 0; d < D_; ++d) acc += hv[i][d] * hv[j][d] * a[d];
      v = leaky_(acc);
    }
    att[i][j] = v;
  }
  __syncthreads();
  if (threadIdx.x < L_) {
    int i = threadIdx.x;
    float mx = -1e30f;
    for (int j = 0; j < L_; ++j) mx = fmaxf(mx, att[i][j]);
    float s = 0.f;
    for (int j = 0; j < L_; ++j) { float e = __expf(att[i][j] - mx); att[i][j] = e; s += e; }
    rinv[i] = 1.f / s;
  }
  __syncthreads();
  for (int p = threadIdx.x; p < L_ * D_; p += 256) {
    int i = p >> 7, d = p & 127;
    float acc = 0.f;
    for (int j = 0; j < L_; ++j) acc += att[i][j] * hv[j][d];
    h_local[(long)b * L_ * D_ + p] = acc * rinv[i];
  }
}

// ------------------------------------------------- fused global aggregator ---
// Block = 256 thr = 8 waves; handles 2 m-rows (16 s-rows). Branch-free WMMA
// x@gw1 (K padded to 132 in LDS and in gw1T), fused leakyrelu + dot-gw2 score
// (shfl-reduced), softmax over S, neighbor sum, and w3 matmul.
__global__ void agg_kernel(const float* __restrict__ emb,
                           const float* __restrict__ selfVal, const int* __restrict__ selfIdx,
                           const int* __restrict__ neighIdx, const float* __restrict__ neighVal,
                           const float* __restrict__ neighW,
                           const float* __restrict__ sess,
                           const float* __restrict__ gw1T, const float* __restrict__ gw2p,
                           const float* __restrict__ gw3p,
                           float* __restrict__ out, int M) {
  __shared__ float nv[16][D_];     // neighbor vectors
  __shared__ float xs[16][KP_];    // padded x rows
  __shared__ float selfv[2][D_];
  __shared__ float sessv[D_];
  __shared__ float wgt[16];
  __shared__ float sc[16];
  __shared__ float alpha[16];
  __shared__ float nout[2][D_];

  int tid = threadIdx.x;
  int lane = tid & 31, laneIdx = lane & 15, laneHalf = lane >> 4;
  int wave = tid >> 5;
  int n0 = wave * 16;

  int p = blockIdx.x;
  int b = (2 * p) / M;
  int m0 = (2 * p) % M;
  long base  = (long)b * M + m0;  // self rows base
  long nbase = base * S_;         // neighbor rows base

  if (tid < D_) sessv[tid] = sess[(b << 7) + tid];
  {
    int r = tid >> 7, d = tid & 127;
    float v = selfVal ? selfVal[((base + r) << 7) + d]
                      : emb[((long)selfIdx[base + r] << 7) + d];
    selfv[r][d] = v;
  }
  for (int i = tid; i < 16 * D_; i += 256) {
    int r = i >> 7, d = i & 127;
    float v = neighIdx ? emb[((long)neighIdx[nbase + r] << 7) + d]
                       : neighVal[((nbase + r) << 7) + d];
    nv[r][d] = v;
  }
  if (tid < 16) { wgt[tid] = neighW[nbase + tid]; sc[tid] = 0.f; }
  __syncthreads();

  // build padded x rows: [sess*nv | w | 0 0 0]
  for (int i = tid; i < 16 * KP_; i += 256) {
    int r = i / KP_, k = i % KP_;
    float v = (k < D_) ? sessv[k] * nv[r][k] : (k == D_ ? wgt[r] : 0.f);
    xs[r][k] = v;
  }
  __syncthreads();

  // alpha_mat = leakyrelu(x @ gw1); branch-free K loop, v2f (b64) loads
  v8f c = {};
  const float* bp = gw1T + (n0 + laneIdx) * KP_;
  {
    const float* ap = &xs[laneIdx][0];
    #pragma unroll 3
    for (int k0 = 0; k0 < KP_; k0 += 4) {
      int kk = k0 + (laneHalf << 1);
      v2f a = *(const v2f*)(ap + kk);
      v2f bb = *(const v2f*)(bp + kk);
      c = wmma4(a, bb, c);
    }
  }
  // fused score: sc[row] += sum_n leaky(c)*gw2[n]; shfl-reduce 16-lane halves
  float g2 = gw2p[n0 + laneIdx];
  float* cf = (float*)&c;
  #pragma unroll
  for (int v = 0; v < 8; ++v) {
    float val = leaky_(cf[v]) * g2;
    val += __shfl_xor(val, 1);
    val += __shfl_xor(val, 2);
    val += __shfl_xor(val, 4);
    val += __shfl_xor(val, 8);
    if (laneIdx == 0) atomicAdd(&sc[v + (laneHalf << 3)], val);
  }
  __syncthreads();

  if (tid < 2) {  // softmax over S=8 per m
    float mx = -1e30f;
    for (int s = 0; s < 8; ++s) mx = fmaxf(mx, sc[tid * 8 + s]);
    float ex[8], sum = 0.f;
    for (int s = 0; s < 8; ++s) { ex[s] = __expf(sc[tid * 8 + s] - mx); sum += ex[s]; }
    float inv = 1.f / sum;
    for (int s = 0; s < 8; ++s) alpha[tid * 8 + s] = ex[s] * inv;
  }
  __syncthreads();
  {
    int r = tid >> 7, d = tid & 127;
    float acc = 0.f;
    #pragma unroll
    for (int s = 0; s < 8; ++s) acc += alpha[(r << 3) + s] * nv[(r << 3) + s][d];
    nout[r][d] = acc;
  }
  __syncthreads();
  {
    int r = tid >> 7, d = tid & 127;
    float acc = 0.f;
    for (int e = 0; e < D_; ++e)
      acc += selfv[r][e] * gw3p[(e << 7) + d] + nout[r][e] * gw3p[((D_ + e) << 7) + d];
    out[((base + r) << 7) + d] = fmaxf(acc, 0.f);
  }
}

// -------------------------------------------------- generic WMMA GEMM -------
// C[M,N] = act( A[M,K] @ BT[N,K]^T (+rowBias[row%biasMod]) ), K % 4 == 0.
__global__ void gemm16_kernel(const float* __restrict__ A, int lda,
                              const float* __restrict__ BT, int ldbt,
                              float* __restrict__ C, long ldc,
                              int M, int N, int K,
                              const float* __restrict__ rowBias, int biasMod,
                              int act) {
  int tilesM = (M + 15) >> 4;
  long tilesN = (long)((N + 15) >> 4);
  long total = (long)tilesM * tilesN;
  int lane = threadIdx.x & 31, laneIdx = lane & 15, laneHalf = lane >> 4;
  for (long t = (long)blockIdx.x * 8 + (threadIdx.x >> 5); t < total;
       t += (long)gridDim.x * 8) {
    int m0 = (int)(t % tilesM) << 4;
    long n0 = (t / tilesM) << 4;
    int row = m0 + laneIdx;
    long col = n0 + laneIdx;
    v8f c = {};
    bool full = (m0 + 16 <= M) && (n0 + 16 <= N);   // uniform per wave
    if (full) {
      const float* ap = A + (long)row * lda;
      const float* bp = BT + col * ldbt;
      for (int k0 = 0; k0 < K; k0 += 4) {
        int kk = k0 + (laneHalf << 1);
        v2f a = *(const v2f*)(ap + kk);
        v2f b = *(const v2f*)(bp + kk);
        c = wmma4(a, b, c);
      }
    } else {
      for (int k0 = 0; k0 < K; k0 += 4) {
        int kk = k0 + (laneHalf << 1);
        v2f a, b;
        a.x = (row < M) ? A[(long)row * lda + kk] : 0.f;
        a.y = (row < M) ? A[(long)row * lda + kk + 1] : 0.f;
        b.x = (col < N) ? BT[col * ldbt + kk] : 0.f;
        b.y = (col < N) ? BT[col * ldbt + kk + 1] : 0.f;
        c = wmma4(a, b, c);
      }
    }
    float* cf = (float*)&c;
    #pragma unroll
    for (int v = 0; v < 8; ++v) {
      int rr = m0 + v + (laneHalf << 3);
      if (rr < M && col < N) {
        float val = cf[v];
        if (rowBias) val += rowBias[(long)(rr % biasMod) * N + col];
        if (act == 1) val = tanhf(val);
        C[(long)rr * ldc + col] = val;
      }
    }
  }
}

// ----------------------------------------------------------- small ops ------
__global__ void add_kernel(const float* a, const float* b, float* o, int n) {
  int i = blockIdx.x * 256 + threadIdx.x;
  if (i < n) o[i] = a[i] + b[i];
}

__global__ void zero1_kernel(float* p) { if (threadIdx.x == 0 && blockIdx.x == 0) p[0] = 0.f; }

__global__ void ssl_kernel(const float* __restrict__ hl, const float* __restrict__ hg,
                           float* __restrict__ loss) {
  int idx = blockIdx.x * 256 + threadIdx.x;  // 16384 (b,d) pairs
  int b = idx >> 7, d = idx & 127;
  int pb = (b * 73 + 21) & 127;              // deterministic stand-in permutations
  float pos = 0.f, neg = 0.f;
  for (int l = 0; l < L_; ++l) {
    int pl = (l * 13 + 7) % L_;
    pos += hl[((b * L_ + l) << 7) + d] * hg[((b * L_ + l) << 7) + d];
    neg += hg[((b * L_ + l) << 7) + d] * hl[((pb * L_ + pl) << 7) + d];
  }
  float term = -__logf(1e-8f + sigmoid_(pos)) - __logf(1e-8f + 1.f - sigmoid_(neg));
  __shared__ float red[256];
  red[threadIdx.x] = term;
  __syncthreads();
  for (int s = 128; s > 0; s >>= 1) {
    if (threadIdx.x < s) red[threadIdx.x] += red[threadIdx.x + s];
    __syncthreads();
  }
  if (threadIdx.x == 0) atomicAdd(loss, BETA_ * red[0]);
}

__global__ void hs_kernel(const float* __restrict__ hidden, const int* __restrict__ mask,
                          float* __restrict__ hs) {
  int b = blockIdx.x, d = threadIdx.x;
  float acc = 0.f, ms = 0.f;
  for (int l = 0; l < L_; ++l) {
    float m = (float)mask[b * L_ + l];
    ms += m;
    acc += hidden[((b * L_ + l) << 7) + d] * m;
  }
  hs[(b << 7) + d] = acc / ms;
}

__global__ void hsg_kernel(const float* __restrict__ hs, const float* __restrict__ glu2_w,
                           float* __restrict__ hsg) {
  __shared__ float hr[D_];
  int b = blockIdx.x, d = threadIdx.x;
  hr[d] = hs[(b << 7) + d];
  __syncthreads();
  float acc = 0.f;
  for (int e = 0; e < D_; ++e) acc += hr[e] * glu2_w[(d << 7) + e];  // glu2_w.T
  hsg[(b << 7) + d] = acc;
}

__global__ void pos_kernel(const float* __restrict__ pos_emb, const float* __restrict__ w1,
                           float* __restrict__ P) {
  __shared__ float pr[D_];
  int l = blockIdx.x, d = threadIdx.x;
  pr[d] = pos_emb[(l << 7) + d];
  __syncthreads();
  float acc = 0.f;
  for (int e = 0; e < D_; ++e) acc += pr[e] * w1[(e << 7) + d];  // top half of w1
  P[(l << 7) + d] = acc;
}

__global__ void beta_kernel(const float* __restrict__ nh, const float* __restrict__ glu1_w,
                            const float* __restrict__ glu1_b, const float* __restrict__ hsg,
                            const float* __restrict__ w2, const int* __restrict__ mask,
                            float* __restrict__ beta) {
  __shared__ float nrow[D_];
  __shared__ float red[D_];
  int row = blockIdx.x;       // 0..B*L-1
  int b = row / L_;
  int d = threadIdx.x;
  nrow[d] = nh[((long)row << 7) + d];
  __syncthreads();
  float acc = glu1_b[d] + hsg[(b << 7) + d];
  for (int e = 0; e < D_; ++e) acc += nrow[e] * glu1_w[(d << 7) + e];  // glu1_w.T
  red[d] = sigmoid_(acc) * w2[d];
  __syncthreads();
  for (int s = 64; s > 0; s >>= 1) {
    if (d < s) red[d] += red[d + s];
    __syncthreads();
  }
  if (d == 0) beta[row] = red[0] * (float)mask[row];
}

__global__ void select_kernel(const float* __restrict__ hidden, const float* __restrict__ beta,
                              float* __restrict__ sel) {
  int b = blockIdx.x, d = threadIdx.x;
  float acc = 0.f;
  for (int l = 0; l < L_; ++l) acc += beta[b * L_ + l] * hidden[((b * L_ + l) << 7) + d];
  sel[(b << 7) + d] = acc;
}

__global__ void norm_kernel(const float* __restrict__ sel, float* __restrict__ snorm) {
  __shared__ float red[D_];
  int b = blockIdx.x, d = threadIdx.x;
  float v = sel[(b << 7) + d];
  red[d] = v * v + 1e-6f;
  __syncthreads();
  for (int s = 64; s > 0; s >>= 1) {
    if (d < s) red[d] += red[d + s];
    __syncthreads();
  }
  if (d == 0) snorm[b] = sqrtf(red[0]);
}

__global__ void fn_kernel(const float* __restrict__ sel, const float* __restrict__ snorm,
                          float* __restrict__ selF) {
  __shared__ float row[B_];
  __shared__ float selfrow[D_];
  __shared__ float tvs[9];
  __shared__ int tis[9];
  int i = blockIdx.x, t = threadIdx.x;
  selfrow[t] = sel[(i << 7) + t];
  __syncthreads();
  float acc = 0.f;
  for (int d = 0; d < D_; ++d) acc += selfrow[d] * sel[(t << 7) + d];
  row[t] = acc / (snorm[i] * snorm[t]);
  __syncthreads();
  if (t == 0) {
    float mx = -1e30f;
    for (int j = 0; j < B_; ++j) mx = fmaxf(mx, row[j]);
    float s = 0.f;
    for (int j = 0; j < B_; ++j) { row[j] = __expf(row[j] - mx); s += row[j]; }
    float inv = 1.f / s;
    for (int j = 0; j < B_; ++j) row[j] *= inv;
    for (int q = 0; q < 9; ++q) {   // top-9, sorted desc
      float bm = -1.f; int bi = 0;
      for (int j = 0; j < B_; ++j) if (row[j] > bm) { bm = row[j]; bi = j; }
      tvs[q] = bm; tis[q] = bi; row[bi] = -2.f;
    }
    float mx2 = tvs[0], s2 = 0.f;
    for (int q = 0; q < 9; ++q) { tvs[q] = __expf(tvs[q] - mx2); s2 += tvs[q]; }
    float inv2 = 1.f / s2;
    for (int q = 0; q < 9; ++q) tvs[q] *= inv2;
  }
  __syncthreads();
  float nb = 0.f;
  for (int q = 0; q < 9; ++q) nb += tvs[q] * sel[(tis[q] << 7) + t];
  selF[(i << 7) + t] = selfrow[t] + nb;
}

// ----------------------------------------------------------------------------
extern "C" void kernel_launch(void* const* d_in, const int* in_sizes, int n_in,
                              void* d_out, int out_size, void* d_ws, size_t ws_size,
                              hipStream_t stream) {
  (void)in_sizes; (void)n_in; (void)out_size; (void)ws_size;
  const int*   inputs  = (const int*)d_in[0];
  const int*   adj     = (const int*)d_in[1];
  const int*   mask    = (const int*)d_in[2];
  const int*   item    = (const int*)d_in[3];
  const int*   adj_all = (const int*)d_in[5];
  const float* num_w   = (const float*)d_in[6];
  const float* emb     = (const float*)d_in[7];
  const float* pos_emb = (const float*)d_in[8];
  const float* a_local = (const float*)d_in[9];
  const float* gw1     = (const float*)d_in[10];
  const float* gw2     = (const float*)d_in[11];
  const float* gw3     = (const float*)d_in[12];
  const float* w1      = (const float*)d_in[13];
  const float* w2      = (const float*)d_in[14];
  const float* glu1_w  = (const float*)d_in[15];
  const float* glu1_b  = (const float*)d_in[16];
  const float* glu2_w  = (const float*)d_in[17];

  float* out0   = (float*)d_out;            // [B,L,D] = h_local + h_global
  float* lossP  = out0 + B_ * L_ * D_;      // scalar
  float* scores = lossP + 1;                // [B, NN-1]

  char* wp = (char*)d_ws;
  auto alloc = [&](size_t nfloats) -> float* {
    float* r = (float*)wp;
    wp += ((nfloats * sizeof(float) + 255) / 256) * 256;
    return r;
  };
  float* h        = alloc(B_ * L_ * D_);
  float* h_local  = alloc(B_ * L_ * D_);
  float* h_global = alloc(B_ * L_ * D_);
  float* sessb    = alloc(B_ * D_);
  float* e0p      = alloc(B_ * L_ * D_);
  float* e1p      = alloc((size_t)B_ * 400 * D_);
  float* hsb      = alloc(B_ * D_);
  float* hsgb     = alloc(B_ * D_);
  float* Pb       = alloc(L_ * D_);
  float* nh       = alloc(B_ * L_ * D_);
  float* betab    = alloc(B_ * L_);
  float* selb     = alloc(B_ * D_);
  float* selF     = alloc(B_ * D_);
  float* snorm    = alloc(B_);
  float* gw1T     = alloc(2 * D_ * KP_);
  float* w1botT   = alloc(D_ * D_);
  float* wn1      = alloc(B_ * 400);
  float* wn2      = alloc((size_t)B_ * 3200);
  int*   n1       = (int*)alloc(B_ * 400);
  int*   n2       = (int*)alloc((size_t)B_ * 3200);

  // 0) weight prep (pad + transpose)
  gw1t_kernel<<<dim3((2 * D_ * KP_ + 255) / 256), 256, 0, stream>>>(gw1, gw1T);
  w1t_kernel<<<dim3((D_ * D_ + 255) / 256), 256, 0, stream>>>(w1, w1botT);
  // 1) h = embedding[inputs]
  gather_h_kernel<<<dim3((B_ * L_ * D_) / 256), 256, 0, stream>>>(emb, inputs, h, B_ * L_ * D_);
  // 2) session mean
  sess_kernel<<<dim3(B_), dim3(D_), 0, stream>>>(emb, item, mask, sessb);
  // 3) neighbor sampling (2 hops)
  hop_kernel<<<dim3((B_ * L_ * S_ + 255) / 256), 256, 0, stream>>>(inputs, adj_all, num_w, n1, wn1, B_ * L_ * S_);
  hop_kernel<<<dim3((B_ * 400 * S_ + 255) / 256), 256, 0, stream>>>(n1, adj_all, num_w, n2, wn2, B_ * 400 * S_);
  // 4) local attention aggregation
  local_agg_kernel<<<dim3(B_), 256, 0, stream>>>(h, adj, a_local, h_local);
  // 5) global aggregation: round 0 (hop weights 0), round 1 (hop weights 1)
  agg_kernel<<<dim3(B_ * L_ / 2), 256, 0, stream>>>(emb, h, nullptr, n1, nullptr, wn1,
      sessb, gw1T, gw2, gw3, e0p, L_);
  agg_kernel<<<dim3(B_ * 400 / 2), 256, 0, stream>>>(emb, nullptr, n1, n2, nullptr, wn2,
      sessb, gw1T, gw2, gw3, e1p, 400);
  agg_kernel<<<dim3(B_ * L_ / 2), 256, 0, stream>>>(emb, e0p, nullptr, nullptr, e1p, wn1,
      sessb, gw1T + D_ * KP_, gw2 + D_, gw3 + 2 * D_ * D_, h_global, L_);
  // 6) output = h_local + h_global (first d_out region; also "hidden" below)
  add_kernel<<<dim3((B_ * L_ * D_) / 256), 256, 0, stream>>>(h_local, h_global, out0, B_ * L_ * D_);
  // 7) SSL loss scalar
  zero1_kernel<<<1, 1, 0, stream>>>(lossP);
  ssl_kernel<<<dim3(64), 256, 0, stream>>>(h_local, h_global, lossP);
  // 8) score head
  hs_kernel<<<dim3(B_), dim3(D_), 0, stream>>>(out0, mask, hsb);
  hsg_kernel<<<dim3(B_), dim3(D_), 0, stream>>>(hsb, glu2_w, hsgb);
  pos_kernel<<<dim3(L_), dim3(D_), 0, stream>>>(pos_emb, w1, Pb);
  {  // nh = tanh(hidden @ w1[128:,:] + P[l])   (BT = w1botT [N=128][K=128])
    int tiles = ((B_ * L_ + 15) / 16) * ((D_ + 15) / 16);
    gemm16_kernel<<<dim3((tiles + 7) / 8), 256, 0, stream>>>(
        out0, D_, w1botT, D_, nh, D_, B_ * L_, D_, D_, Pb, L_, 1);
  }
  beta_kernel<<<dim3(B_ * L_), dim3(D_), 0, stream>>>(nh, glu1_w, glu1_b, hsgb, w2, mask, betab);
  select_kernel<<<dim3(B_), dim3(D_), 0, stream>>>(out0, betab, selb);
  norm_kernel<<<dim3(B_), dim3(D_), 0, stream>>>(selb, snorm);
  fn_kernel<<<dim3(B_), dim3(D_), 0, stream>>>(selb, snorm, selF);
  {  // scores = selF @ embedding[1:].T  (BT = emb+D_, naturally [N][K])
    int Nsc = NN_ - 1;
    long tiles = (long)((B_ + 15) / 16) * ((Nsc + 15) / 16);
    gemm16_kernel<<<dim3((unsigned)((tiles + 7) / 8)), 256, 0, stream>>>(
        selF, D_, emb + D_, D_, scores, (long)Nsc, B_, Nsc, D_, nullptr, 1, 0);
  }
}